// NeuralSolver_12378095747571
// MI455X (gfx1250) — compile-verified
//
#include <hip/hip_runtime.h>
#include <hip/hip_bf16.h>
#include <math.h>

// Problem constants (from reference)
#define B_      8
#define NP_     81920
#define DL_     16
#define DA_     4
#define DTOT_   20            // DL + DA
#define H_      128
#define NB_     3
#define FANIN_  60            // NB * DTOT
#define DT_     0.1f
#define WAVES_  8             // waves per block
#define TILE_   16            // points per wave (WMMA M)

typedef __attribute__((ext_vector_type(2))) float v2f;
typedef __attribute__((ext_vector_type(8))) float v8f;

// Hardware transcendental tanh (V_TANH_F32, new on gfx1250). Branch-free,
// single TRANS32 op that can co-execute with the WMMA pipe, unlike the
// branchy OCML tanhf expansion.
__device__ __forceinline__ float fast_tanh(float x) {
#if __has_builtin(__builtin_amdgcn_tanhf)
    return __builtin_amdgcn_tanhf(x);
#else
    float r;
    // v_nop satisfies the TRANS-op result-use hazard (1 independent
    // instruction required) since the compiler can't see into the asm.
    asm volatile("v_tanh_f32 %0, %1\n\tv_nop" : "=v"(r) : "v"(x));
    return r;
#endif
}

// One forward-Euler step: Yout = Yin with Yout[..., :DL] += DT * Phi(gather(Yin))
__global__ __launch_bounds__(256) void NeuralSolver_step(
    const float* __restrict__ Yin,
    float* __restrict__ Yout,
    const long long* __restrict__ nbr,   // (NP, NB) int64
    const float* __restrict__ W1,        // (60, 128)
    const float* __restrict__ b1,        // (128,)
    const float* __restrict__ W2,        // (128, 16)
    const float* __restrict__ b2)        // (16,)
{
    // LDS: weights pre-swizzled into WMMA B-fragment lane order, plus a
    // 16x16 h-tile per wave for the GEMM1->GEMM2 transpose.
    __shared__ float w1s[15 * 8 * 32 * 2];   // [kt][nt][lane][r]  = 7680 f
    __shared__ float w2s[32 * 32 * 2];       // [ktg][lane][r]     = 2048 f
    __shared__ float b1s[H_];
    __shared__ float b2s[DL_];
    __shared__ float htile[WAVES_ * 256];    // per-wave 16x16 tile

    const int tid = threadIdx.x;

    // ---- swizzle W1 into B-fragment layout ----
    // B-frag of V_WMMA_F32_16X16X4_F32 (4x16): VGPR r, lanes 0-15 -> K = r,
    // lanes 16-31 -> K = 2 + r (mirrors documented A 16x4 layout).
    for (int idx = tid; idx < 15 * 8 * 32 * 2; idx += 256) {
        int r    = idx & 1;
        int lane = (idx >> 1) & 31;
        int nt   = (idx >> 6) & 7;
        int kt   = idx >> 9;
        int k    = 4 * kt + 2 * (lane >> 4) + r;          // 0..59
        int col  = nt * 16 + (lane & 15);                 // 0..127
        w1s[idx] = W1[k * H_ + col];
    }
    for (int idx = tid; idx < 32 * 32 * 2; idx += 256) {
        int r    = idx & 1;
        int lane = (idx >> 1) & 31;
        int ktg  = idx >> 6;
        int k    = 4 * ktg + 2 * (lane >> 4) + r;         // 0..127
        w2s[idx] = W2[k * DL_ + (lane & 15)];
    }
    if (tid < H_)  b1s[tid] = b1[tid];
    if (tid < DL_) b2s[tid] = b2[tid];
    __syncthreads();

    const int l    = tid & 31;          // lane
    const int wave = tid >> 5;
    const int half = l >> 4;            // 0: K%4 in {0,1}; 1: K%4 in {2,3}
    const int m    = l & 15;            // row / column id within 16
    const int b    = blockIdx.y;
    const int p0   = (blockIdx.x * WAVES_ + wave) * TILE_;
    const int p    = p0 + m;            // this lane's gather point

    // ---- gather A fragments (16 points x 60 features, fp32) ----
    long long n0 = nbr[(size_t)p * NB_ + 0];
    long long n1 = nbr[(size_t)p * NB_ + 1];
    long long n2 = nbr[(size_t)p * NB_ + 2];
    const size_t rowbase = (size_t)b * NP_;

    v2f afrag[15];
#pragma unroll
    for (int kt = 0; kt < 15; ++kt) {
        int k0 = 4 * kt + 2 * half;                  // even; never straddles a neighbor
        int j  = k0 / DTOT_;
        int c  = k0 - DTOT_ * j;
        long long n = (j == 0) ? n0 : (j == 1) ? n1 : n2;
        const float* src = Yin + (rowbase + (size_t)n) * DTOT_ + c;
        afrag[kt] = *(const v2f*)src;                // 8B-aligned global_load_b64
    }

    // ---- GEMM2 accumulator seeded with b2 (C layout: N = lane%16) ----
    v8f acc2;
    {
        float bv = b2s[m];
#pragma unroll
        for (int i = 0; i < 8; ++i) acc2[i] = bv;
    }

    // ---- fused GEMM1 (tanh) -> GEMM2, one H n-tile at a time ----
#pragma unroll 1
    for (int nt = 0; nt < 8; ++nt) {
        v8f acc1;
        {
            float bv = b1s[nt * 16 + m];
#pragma unroll
            for (int i = 0; i < 8; ++i) acc1[i] = bv;
        }
#pragma unroll
        for (int kt = 0; kt < 15; ++kt) {
            v2f bf = *(const v2f*)&w1s[(((kt << 3) + nt) << 6) + (l << 1)];
            acc1 = __builtin_amdgcn_wmma_f32_16x16x4_f32(
                false, afrag[kt], false, bf, (short)0, acc1, false, false);
        }
        // tanh and store the 16x16 h-tile to LDS in [row][col] order
#pragma unroll
        for (int i = 0; i < 8; ++i) {
            int M = i + (half << 3);
            htile[(wave << 8) + (M << 4) + m] = fast_tanh(acc1[i]);
        }
        asm volatile("s_wait_dscnt 0" ::: "memory");  // cross-lane store->load
        // this h-tile is exactly K-chunk [nt*16, nt*16+16) of GEMM2
#pragma unroll
        for (int kt2 = 0; kt2 < 4; ++kt2) {
            v2f a2  = *(const v2f*)&htile[(wave << 8) + (m << 4) + 4 * kt2 + 2 * half];
            v2f bf2 = *(const v2f*)&w2s[((((nt << 2) + kt2)) << 6) + (l << 1)];
            acc2 = __builtin_amdgcn_wmma_f32_16x16x4_f32(
                false, a2, false, bf2, (short)0, acc2, false, false);
        }
    }

    // ---- epilogue: Yout[:, :, :16] = Yin + DT * fZ ; copy aux channels ----
#pragma unroll
    for (int r = 0; r < 8; ++r) {
        int M = r + (half << 3);
        size_t idx = (rowbase + (size_t)(p0 + M)) * DTOT_ + m;
        Yout[idx] = Yin[idx] + DT_ * acc2[r];
    }
    {
        int pm = l >> 1;
        int c  = DL_ + ((l & 1) << 1);
        size_t idx = (rowbase + (size_t)(p0 + pm)) * DTOT_ + c;
        Yout[idx]     = Yin[idx];
        Yout[idx + 1] = Yin[idx + 1];
    }
}

extern "C" void kernel_launch(void* const* d_in, const int* in_sizes, int n_in,
                              void* d_out, int out_size, void* d_ws, size_t ws_size,
                              hipStream_t stream) {
    (void)in_sizes; (void)n_in; (void)out_size; (void)ws_size;
    const float*     Y0  = (const float*)d_in[0];
    const long long* nbr = (const long long*)d_in[1];
    const float*     W1  = (const float*)d_in[2];
    const float*     b1  = (const float*)d_in[3];
    const float*     W2  = (const float*)d_in[4];
    const float*     b2  = (const float*)d_in[5];
    float* out = (float*)d_out;
    float* ws  = (float*)d_ws;   // one full Y copy (52.4 MB) for ping-pong

    dim3 grid(NP_ / (TILE_ * WAVES_), B_);   // (640, 8)
    dim3 block(256);

    // 4 Euler steps with a global dependency each -> 4 launches, ping-pong.
    NeuralSolver_step<<<grid, block, 0, stream>>>(Y0,  ws,  nbr, W1, b1, W2, b2);
    NeuralSolver_step<<<grid, block, 0, stream>>>(ws,  out, nbr, W1, b1, W2, b2);
    NeuralSolver_step<<<grid, block, 0, stream>>>(out, ws,  nbr, W1, b1, W2, b2);
    NeuralSolver_step<<<grid, block, 0, stream>>>(ws,  out, nbr, W1, b1, W2, b2);
}